// SGRUCell_89696097009943
// MI455X (gfx1250) — compile-verified
//
#include <hip/hip_runtime.h>
#include <math.h>

#define B_  64
#define D_  256
#define H_  512
#define R_  32
#define HR_ 544
#define CLIPV 50.0f

typedef float v2f __attribute__((ext_vector_type(2)));
typedef float v8f __attribute__((ext_vector_type(8)));
typedef float f4  __attribute__((ext_vector_type(4)));

// ---------------------------------------------------------------------------
// Kernel 1: gates = x @ x2h_w.T + x2h_b + h @ h2h_w.T + h2h_b   -> (64, 544)
// One 16x16 output tile per wave using V_WMMA_F32_16X16X4_F32 (fp32 WMMA).
// A (16x4 f32): lanes 0-15 = M rows, VGPR0/1 = K pair; lanes 16-31 = K+2 pair.
// B (4x16 f32): mirrored with N on lanes. C/D: VGPR v -> M = v + 8*(lane/16).
// ---------------------------------------------------------------------------
__global__ __launch_bounds__(128) void gates_gemm_wmma(
    const float* __restrict__ x, const float* __restrict__ h,
    const float* __restrict__ x2h_w, const float* __restrict__ x2h_b,
    const float* __restrict__ h2h_w, const float* __restrict__ h2h_b,
    float* __restrict__ gates)
{
    const int lane  = threadIdx.x & 31;
    const int wave  = threadIdx.x >> 5;
    const int n0    = blockIdx.x * 16;     // 34 N-tiles
    const int m0    = wave * 16;           // 4 M-tiles (64 rows)
    const int r     = lane & 15;           // A: M row / B,C: N col
    const int khalf = lane >> 4;           // 0 or 1

    v8f c = {};

    // x @ x2h_w.T : K = D = 256
    for (int k = 0; k < D_; k += 4) {
        const int kk = k + 2 * khalf;
        v2f a = *(const v2f*)(x     + (size_t)(m0 + r) * D_ + kk);
        v2f b = *(const v2f*)(x2h_w + (size_t)(n0 + r) * D_ + kk);
        c = __builtin_amdgcn_wmma_f32_16x16x4_f32(false, a, false, b,
                                                  (short)0, c, false, false);
    }
    // h @ h2h_w.T : K = H = 512
    for (int k = 0; k < H_; k += 4) {
        const int kk = k + 2 * khalf;
        v2f a = *(const v2f*)(h     + (size_t)(m0 + r) * H_ + kk);
        v2f b = *(const v2f*)(h2h_w + (size_t)(n0 + r) * H_ + kk);
        c = __builtin_amdgcn_wmma_f32_16x16x4_f32(false, a, false, b,
                                                  (short)0, c, false, false);
    }

    const int   n    = n0 + r;
    const float bias = x2h_b[n] + h2h_b[n];
#pragma unroll
    for (int v = 0; v < 8; ++v) {
        const int m = m0 + v + 8 * khalf;
        gates[(size_t)m * HR_ + n] = c[v] + bias;
    }
}

// ---------------------------------------------------------------------------
// Kernel 2: plastic[b,i] = sum_j relu(alpha[i,j]) * dU[b,i,j] * h[b,j]
// One wave per (b,i) row; 16 floats/lane via float4; shfl_xor reduce.
// dU loaded with default (RT) policy so it parks in the 192MB L2 for kernel 5.
// ---------------------------------------------------------------------------
__global__ __launch_bounds__(256) void plastic_kernel(
    const float* __restrict__ dU, const float* __restrict__ alpha,
    const float* __restrict__ h, float* __restrict__ plastic)
{
    const int lane = threadIdx.x & 31;
    const int wave = threadIdx.x >> 5;
    const int row  = blockIdx.x * 8 + wave;     // b*512 + i, 0..32767
    const int b    = row >> 9;
    const int i    = row & (H_ - 1);

    const float* dUr = dU    + (size_t)row * H_;
    const float* ar  = alpha + (size_t)i   * H_;
    const float* hr  = h     + (size_t)b   * H_;

    float acc = 0.0f;
#pragma unroll
    for (int t = 0; t < 4; ++t) {
        const int j = (t * 32 + lane) * 4;
        f4 d  = *(const f4*)(dUr + j);
        f4 a  = *(const f4*)(ar  + j);
        f4 hv = *(const f4*)(hr  + j);
        acc += fmaxf(a.x, 0.0f) * d.x * hv.x;
        acc += fmaxf(a.y, 0.0f) * d.y * hv.y;
        acc += fmaxf(a.z, 0.0f) * d.z * hv.z;
        acc += fmaxf(a.w, 0.0f) * d.w * hv.w;
    }
#pragma unroll
    for (int off = 16; off > 0; off >>= 1)
        acc += __shfl_xor(acc, off, 32);
    if (lane == 0) plastic[row] = acc;
}

// ---------------------------------------------------------------------------
// Kernel 3: mod[b] = sum_r relu(gates[b, H + r])
// ---------------------------------------------------------------------------
__global__ void mod_kernel(const float* __restrict__ gates,
                           float* __restrict__ modb)
{
    const int b = blockIdx.x * blockDim.x + threadIdx.x;
    if (b >= B_) return;
    float s = 0.0f;
#pragma unroll
    for (int r = 0; r < R_; ++r)
        s += fmaxf(gates[(size_t)b * HR_ + H_ + r], 0.0f);
    modb[b] = s;
}

// ---------------------------------------------------------------------------
// Kernel 4: v_new, new_h, new_trace_e  (B*H elements)
// ---------------------------------------------------------------------------
__global__ __launch_bounds__(256) void pointwise_kernel(
    const float* __restrict__ gates, const float* __restrict__ plastic,
    const float* __restrict__ v, const float* __restrict__ h,
    const float* __restrict__ trace_e,
    const float* __restrict__ tau_v, const float* __restrict__ tau_e,
    float* __restrict__ v_new, float* __restrict__ new_h,
    float* __restrict__ new_trace_e)
{
    const int idx = blockIdx.x * 256 + threadIdx.x;   // 0..32767
    const int b = idx >> 9;
    const int i = idx & (H_ - 1);

    const float sv = 1.0f / (1.0f + expf(-tau_v[i]));
    const float se = 1.0f / (1.0f + expf(-tau_e[i]));

    const float dv = gates[(size_t)b * HR_ + i] + plastic[idx];
    const float vn = (1.0f - sv) * v[idx] + sv * dv;
    v_new[idx] = vn;
    new_h[idx] = fmaxf(vn, 0.0f);
    new_trace_e[idx] = (1.0f - se) * trace_e[idx] + se * h[idx];
}

// ---------------------------------------------------------------------------
// Kernel 5: the HBM-bound fused pass over (B, H, H):
//   outer      = new_h[b,i]*trace_e_old[b,j] - new_trace_e[b,i]*h[b,j]
//   trace_E'   = (1-sE)*trace_E + sE*outer                     (NT load+store)
//   dU'        = clip((1-sU)*dU + sU*mod[b]*trace_E', lo, hi)  (NT store)
// float4 per thread; alpha / h2h_w stay hot in L2 (reused 64x over b).
// ---------------------------------------------------------------------------
__global__ __launch_bounds__(256) void update_big_kernel(
    const float* __restrict__ dU, const float* __restrict__ trace_E,
    const float* __restrict__ alpha, const float* __restrict__ h2h_w,
    const float* __restrict__ h, const float* __restrict__ trace_e_old,
    const float* __restrict__ new_h, const float* __restrict__ new_trace_e,
    const float* __restrict__ modb,
    const float* __restrict__ tau_U, const float* __restrict__ tau_E,
    float* __restrict__ dU_out, float* __restrict__ trace_E_out)
{
    const size_t flat = (size_t)blockIdx.x * 256 + threadIdx.x; // B*H*(H/4)
    const int    jv   = (int)(flat & 127);
    const size_t row  = flat >> 7;              // b*512 + i
    const int    i    = (int)(row & (H_ - 1));
    const int    b    = (int)(row >> 9);
    const int    j    = jv * 4;

    const float sU = 1.0f / (1.0f + expf(-tau_U[0]));
    const float sE = 1.0f / (1.0f + expf(-tau_E[0]));

    const float nh  = new_h[row];
    const float nte = new_trace_e[row];
    const float mo  = modb[b];

    const size_t big = (row << 9) + j;          // (b*H + i)*H + j
    const f4 te = *(const f4*)(trace_e_old + (size_t)b * H_ + j);
    const f4 hv = *(const f4*)(h           + (size_t)b * H_ + j);
    const f4 tE = __builtin_nontemporal_load((const f4*)(trace_E + big));
    const f4 du = *(const f4*)(dU + big);       // hope for L2 hit (pass 2 warmed it)
    const f4 al = *(const f4*)(alpha + (size_t)i * H_ + j);
    const f4 wq = *(const f4*)(h2h_w + (size_t)i * H_ + j);

    f4 outT, outU;
#define SGRU_COMP(c)                                              \
    {                                                             \
        float outer = nh * te.c - nte * hv.c;                     \
        float tEn   = (1.0f - sE) * tE.c + sE * outer;            \
        float dun   = (1.0f - sU) * du.c + sU * mo * tEn;         \
        float ar    = fmaxf(al.c, 0.0f);                          \
        float denom = ar + 1e-8f;                                 \
        float upper = fmaxf(CLIPV - wq.c, 0.0f) / denom;          \
        float lower = -fmaxf(CLIPV + wq.c, 0.0f) / denom;         \
        dun    = fmaxf(fminf(dun, upper), lower);                 \
        outT.c = tEn;                                             \
        outU.c = dun;                                             \
    }
    SGRU_COMP(x) SGRU_COMP(y) SGRU_COMP(z) SGRU_COMP(w)
#undef SGRU_COMP

    __builtin_nontemporal_store(outT, (f4*)(trace_E_out + big));
    __builtin_nontemporal_store(outU, (f4*)(dU_out + big));
}

// ---------------------------------------------------------------------------
extern "C" void kernel_launch(void* const* d_in, const int* in_sizes, int n_in,
                              void* d_out, int out_size, void* d_ws, size_t ws_size,
                              hipStream_t stream)
{
    const float* x       = (const float*)d_in[0];
    const float* h       = (const float*)d_in[1];
    const float* v       = (const float*)d_in[2];
    const float* dU      = (const float*)d_in[3];
    const float* trace_e = (const float*)d_in[4];
    const float* trace_E = (const float*)d_in[5];
    const float* x2h_w   = (const float*)d_in[6];
    const float* x2h_b   = (const float*)d_in[7];
    const float* h2h_w   = (const float*)d_in[8];
    const float* h2h_b   = (const float*)d_in[9];
    const float* alpha   = (const float*)d_in[10];
    const float* tau_v   = (const float*)d_in[11];
    const float* tau_e   = (const float*)d_in[12];
    const float* tau_U   = (const float*)d_in[13];
    const float* tau_E   = (const float*)d_in[14];

    // Outputs concatenated flat: v_new, new_h, dU_new, new_trace_e, new_trace_E
    float* out      = (float*)d_out;
    float* o_v_new  = out;
    float* o_new_h  = out + (size_t)B_ * H_;
    float* o_dU     = out + (size_t)2 * B_ * H_;
    float* o_te     = o_dU + (size_t)B_ * H_ * H_;
    float* o_tE     = o_te + (size_t)B_ * H_;

    // Workspace: gates (64x544), plastic (64x512), mod (64)
    float* ws      = (float*)d_ws;
    float* gates   = ws;
    float* plastic = ws + (size_t)B_ * HR_;
    float* modb    = plastic + (size_t)B_ * H_;

    // 1) WMMA GEMM for all gate pre-activations
    gates_gemm_wmma<<<HR_ / 16, 128, 0, stream>>>(x, h, x2h_w, x2h_b,
                                                  h2h_w, h2h_b, gates);
    // 2) plastic term (also warms L2 with dU)
    plastic_kernel<<<(B_ * H_) / 8, 256, 0, stream>>>(dU, alpha, h, plastic);
    // 3) modulation scalar per batch
    mod_kernel<<<2, 32, 0, stream>>>(gates, modb);
    // 4) v_new / new_h / new_trace_e
    pointwise_kernel<<<(B_ * H_) / 256, 256, 0, stream>>>(
        gates, plastic, v, h, trace_e, tau_v, tau_e, o_v_new, o_new_h, o_te);
    // 5) fused trace_E / dU update (HBM-bound)
    update_big_kernel<<<(size_t)B_ * H_ * (H_ / 4) / 256, 256, 0, stream>>>(
        dU, trace_E, alpha, h2h_w, h, trace_e, o_new_h, o_te, modb,
        tau_U, tau_E, o_dU, o_tE);
}